// ClassBlock_81243601371743
// MI455X (gfx1250) — compile-verified
//
#include <hip/hip_runtime.h>
#include <math.h>

// ---------------------------------------------------------------------------
// Problem dims (from the reference)
// ---------------------------------------------------------------------------
#define BATCH   64
#define SEQ     1025
#define DIMC    1536
#define DINNER  3072
#define DSTATE  64
#define DTRANK  96
#define NBLK    4
#define BLKC    384          // DIMC / NBLK
#define LAMBD   0.01f

typedef __attribute__((ext_vector_type(2))) float v2f;
typedef __attribute__((ext_vector_type(8))) float v8f;

__device__ __forceinline__ float siluf(float x)     { return x / (1.f + __expf(-x)); }
__device__ __forceinline__ float softplusf(float x) { return x > 0.f ? x + log1pf(__expf(-x)) : log1pf(__expf(x)); }
__device__ __forceinline__ float softshrinkf(float x) {
    return x > LAMBD ? x - LAMBD : (x < -LAMBD ? x + LAMBD : 0.f);
}

// ---------------------------------------------------------------------------
// 1) Bulk pass-through copy of x[:, 1:1025, :]  (the bandwidth roofline term)
// ---------------------------------------------------------------------------
__global__ void copy_tail_k(const float4* __restrict__ src, float4* __restrict__ dst) {
    const long per_b = (long)(SEQ - 1) * DIMC / 4;          // 393216 float4 per batch
    const long total = (long)BATCH * per_b;
    const long row_b = (long)SEQ * DIMC / 4;                // full batch row in float4
    for (long i = (long)blockIdx.x * blockDim.x + threadIdx.x; i < total;
         i += (long)gridDim.x * blockDim.x) {
        long b = i / per_b, r = i - b * per_b;
        long off = b * row_b + (DIMC / 4) + r;              // skip CLS token
        dst[off] = src[off];
    }
}

// ---------------------------------------------------------------------------
// 2) LayerNorm over last dim (1536), one block per row, strided source
// ---------------------------------------------------------------------------
__global__ void layernorm_k(const float* __restrict__ src, long src_ld,
                            const float* __restrict__ g, const float* __restrict__ bta,
                            float* __restrict__ dst) {
    const int row = blockIdx.x;
    const float* x = src + (long)row * src_ld;
    __shared__ float s1[256], s2[256];
    float a = 0.f, b = 0.f;
    for (int c = threadIdx.x; c < DIMC; c += 256) { float v = x[c]; a += v; b += v * v; }
    s1[threadIdx.x] = a; s2[threadIdx.x] = b;
    __syncthreads();
    for (int st = 128; st > 0; st >>= 1) {
        if (threadIdx.x < st) { s1[threadIdx.x] += s1[threadIdx.x + st];
                                s2[threadIdx.x] += s2[threadIdx.x + st]; }
        __syncthreads();
    }
    const float mean = s1[0] * (1.f / DIMC);
    const float var  = s2[0] * (1.f / DIMC) - mean * mean;
    const float inv  = rsqrtf(var + 1e-5f);
    for (int c = threadIdx.x; c < DIMC; c += 256)
        dst[(long)row * DIMC + c] = (x[c] - mean) * inv * g[c] + bta[c];
}

// ---------------------------------------------------------------------------
// 3) Generic WMMA f32 GEMM:  C[M,N] = A[M,K] @ W^T   (W is N x K row-major)
//    One wave per 16x16 tile.  grid = (N/16, M/16), block = 32.
//    mode 0: plain   mode 1: softplus(acc + bias[col])   mode 2: acc + resid
// ---------------------------------------------------------------------------
__global__ void gemm_wT_k(const float* __restrict__ A, long lda,
                          const float* __restrict__ W,
                          float* __restrict__ C, long ldc,
                          int K, int mode,
                          const float* __restrict__ bias,
                          const float* __restrict__ resid, long resid_ld) {
    const int lane = threadIdx.x;
    const int n0 = blockIdx.x * 16;
    const int m0 = blockIdx.y * 16;
    const bool hi = lane >= 16;
    const int  mr = lane & 15;
    // A fragment: lanes 0-15 -> K=kb+0,kb+1 ; lanes 16-31 -> K=kb+2,kb+3
    const float* Ap = A + (long)(m0 + mr) * lda + (hi ? 2 : 0);
    // B fragment (= column mr of W^T = row (n0+mr) of W), same K split
    const float* Wp = W + (long)(n0 + mr) * K + (hi ? 2 : 0);
    v8f acc = {};
    for (int kb = 0; kb < K; kb += 4) {
        v2f a, b;
        a.x = Ap[kb]; a.y = Ap[kb + 1];
        b.x = Wp[kb]; b.y = Wp[kb + 1];
        acc = __builtin_amdgcn_wmma_f32_16x16x4_f32(false, a, false, b,
                                                    (short)0, acc, false, false);
    }
    const int col = n0 + mr;
    #pragma unroll
    for (int i = 0; i < 8; ++i) {
        const int row = m0 + i + (hi ? 8 : 0);
        float v = acc[i];
        if (mode == 1)      v = softplusf(v + bias[col]);
        else if (mode == 2) v = v + resid[(long)row * resid_ld + col];
        C[(long)row * ldc + col] = v;
    }
}

// ---------------------------------------------------------------------------
// 4) Mamba elementwise middle: xc = silu(conv_b + xin * conv_w[:,3])
// ---------------------------------------------------------------------------
__global__ void mid_k(const float* __restrict__ xz, const float* __restrict__ conv_w,
                      const float* __restrict__ conv_b, float* __restrict__ xc) {
    int idx = blockIdx.x * blockDim.x + threadIdx.x;
    if (idx >= BATCH * DINNER) return;
    int b = idx / DINNER, d = idx - b * DINNER;
    float xin = xz[(long)b * (2 * DINNER) + d];
    xc[idx] = siluf(conv_b[d] + xin * conv_w[d * 4 + 3]);
}

// ---------------------------------------------------------------------------
// 5) dotBC[b] = sum_s B[b,s] * C[b,s]   (cols 96..159 and 160..223 of dbl)
// ---------------------------------------------------------------------------
__global__ void bc_k(const float* __restrict__ dbl, float* __restrict__ dotBC) {
    __shared__ float s[64];
    int b = blockIdx.x, t = threadIdx.x;
    const float* row = dbl + (long)b * (DTRANK + 2 * DSTATE);
    s[t] = row[DTRANK + t] * row[DTRANK + DSTATE + t];
    __syncthreads();
    for (int st = 32; st > 0; st >>= 1) {
        if (t < st) s[t] += s[t + st];
        __syncthreads();
    }
    if (t == 0) dotBC[b] = s[0];
}

// ---------------------------------------------------------------------------
// 6) y = (dt*dotBC + D) * xc * silu(z)
// ---------------------------------------------------------------------------
__global__ void y_k(const float* __restrict__ dt, const float* __restrict__ dotBC,
                    const float* __restrict__ Dp, const float* __restrict__ xc,
                    const float* __restrict__ xz, float* __restrict__ yact) {
    int idx = blockIdx.x * blockDim.x + threadIdx.x;
    if (idx >= BATCH * DINNER) return;
    int b = idx / DINNER, d = idx - b * DINNER;
    float z = xz[(long)b * (2 * DINNER) + DINNER + d];
    yact[idx] = (dt[idx] * dotBC[b] + Dp[d]) * xc[idx] * siluf(z);
}

// ---------------------------------------------------------------------------
// 7) Analytic length-4 forward FFT (ortho) along the block axis
// ---------------------------------------------------------------------------
__global__ void fftpre_k(const float* __restrict__ ln2,
                         float* __restrict__ Xr, float* __restrict__ Xi) {
    int idx = blockIdx.x * blockDim.x + threadIdx.x;
    if (idx >= BATCH * BLKC) return;
    int b = idx / BLKC, d = idx - b * BLKC;
    const float* p = ln2 + (long)b * DIMC + d;
    float x0 = p[0], x1 = p[BLKC], x2 = p[2 * BLKC], x3 = p[3 * BLKC];
    float* pr = Xr + (long)b * DIMC + d;
    float* pi = Xi + (long)b * DIMC + d;
    float e = 0.5f * (x1 - x3);
    pr[0]        = 0.5f * (x0 + x1 + x2 + x3);  pi[0]        = 0.f;
    pr[BLKC]     = 0.5f * (x0 - x2);            pi[BLKC]     = -e;
    pr[2 * BLKC] = 0.5f * (x0 - x1 + x2 - x3);  pi[2 * BLKC] = 0.f;
    pr[3 * BLKC] = 0.5f * (x0 - x2);            pi[3 * BLKC] = e;
}

// ---------------------------------------------------------------------------
// 8) Fused complex GEMM per FFT block (W is K x N row-major, per block j):
//    R = Ar@W0 - Ai@W1 + b0 ; I = Ar@W1 + Ai@W0 + b1 ; then relu/softshrink
//    grid = (BLKC/16, BATCH/16, NBLK), block = 32 (one wave per tile)
// ---------------------------------------------------------------------------
__global__ void cgemm_k(const float* __restrict__ Ar, const float* __restrict__ Ai,
                        const float* __restrict__ Wc,   // (2, NBLK, BLKC, BLKC)
                        const float* __restrict__ bc,   // (2, NBLK, BLKC)
                        float* __restrict__ Or, float* __restrict__ Oi,
                        int mode) {                      // 0 relu, 1 softshrink
    const int lane = threadIdx.x;
    const int n0 = blockIdx.x * 16;
    const int m0 = blockIdx.y * 16;
    const int j  = blockIdx.z;
    const bool hi = lane >= 16;
    const int  mr = lane & 15;
    const float* ArP = Ar + (long)(m0 + mr) * DIMC + j * BLKC + (hi ? 2 : 0);
    const float* AiP = Ai + (long)(m0 + mr) * DIMC + j * BLKC + (hi ? 2 : 0);
    const float* W0 = Wc + ((long)(0 * NBLK + j)) * BLKC * BLKC;
    const float* W1 = Wc + ((long)(1 * NBLK + j)) * BLKC * BLKC;
    const int col = n0 + mr;
    v8f accR = {}, accI = {};
    for (int kb = 0; kb < BLKC; kb += 4) {
        const int kk = kb + (hi ? 2 : 0);
        v2f ar, ai, nai, b0, b1;
        ar.x = ArP[kb]; ar.y = ArP[kb + 1];
        ai.x = AiP[kb]; ai.y = AiP[kb + 1];
        nai.x = -ai.x;  nai.y = -ai.y;
        b0.x = W0[(long)kk * BLKC + col]; b0.y = W0[(long)(kk + 1) * BLKC + col];
        b1.x = W1[(long)kk * BLKC + col]; b1.y = W1[(long)(kk + 1) * BLKC + col];
        accR = __builtin_amdgcn_wmma_f32_16x16x4_f32(false, ar,  false, b0, (short)0, accR, false, false);
        accR = __builtin_amdgcn_wmma_f32_16x16x4_f32(false, nai, false, b1, (short)0, accR, false, false);
        accI = __builtin_amdgcn_wmma_f32_16x16x4_f32(false, ar,  false, b1, (short)0, accI, false, false);
        accI = __builtin_amdgcn_wmma_f32_16x16x4_f32(false, ai,  false, b0, (short)0, accI, false, false);
    }
    const float br = bc[(0 * NBLK + j) * BLKC + col];
    const float bi = bc[(1 * NBLK + j) * BLKC + col];
    #pragma unroll
    for (int i = 0; i < 8; ++i) {
        const int row = m0 + i + (hi ? 8 : 0);
        float r = accR[i] + br, im = accI[i] + bi;
        if (mode == 0) { r = fmaxf(r, 0.f); im = fmaxf(im, 0.f); }
        else           { r = softshrinkf(r); im = softshrinkf(im); }
        Or[(long)row * DIMC + j * BLKC + col] = r;
        Oi[(long)row * DIMC + j * BLKC + col] = im;
    }
}

// ---------------------------------------------------------------------------
// 9) Analytic length-4 inverse FFT (ortho, real part) + residual -> out[:,0,:]
// ---------------------------------------------------------------------------
__global__ void fftpost_k(const float* __restrict__ r2, const float* __restrict__ i2,
                          const float* __restrict__ cls1, float* __restrict__ out) {
    int idx = blockIdx.x * blockDim.x + threadIdx.x;
    if (idx >= BATCH * BLKC) return;
    int b = idx / BLKC, d = idx - b * BLKC;
    const float* pr = r2 + (long)b * DIMC + d;
    const float* pi = i2 + (long)b * DIMC + d;
    float r0 = pr[0], r1 = pr[BLKC], rr2 = pr[2 * BLKC], r3 = pr[3 * BLKC];
    float i1 = pi[BLKC], i3 = pi[3 * BLKC];
    float y0 = 0.5f * (r0 + r1 + rr2 + r3);
    float y1 = 0.5f * (r0 - i1 - rr2 + i3);
    float y2 = 0.5f * (r0 - r1 + rr2 - r3);
    float y3 = 0.5f * (r0 + i1 - rr2 - i3);
    const float* c = cls1 + (long)b * DIMC + d;
    float* o = out + (long)b * SEQ * DIMC + d;
    o[0]        = c[0]        + y0;
    o[BLKC]     = c[BLKC]     + y1;
    o[2 * BLKC] = c[2 * BLKC] + y2;
    o[3 * BLKC] = c[3 * BLKC] + y3;
}

// ---------------------------------------------------------------------------
// Host launcher
// ---------------------------------------------------------------------------
extern "C" void kernel_launch(void* const* d_in, const int* in_sizes, int n_in,
                              void* d_out, int out_size, void* d_ws, size_t ws_size,
                              hipStream_t stream) {
    const float* x          = (const float*)d_in[0];
    const float* norm1_g    = (const float*)d_in[3];
    const float* norm1_b    = (const float*)d_in[4];
    const float* in_proj_w  = (const float*)d_in[5];   // (6144, 1536)
    const float* conv_w     = (const float*)d_in[6];   // (3072, 4)
    const float* conv_b     = (const float*)d_in[7];
    const float* x_proj_w   = (const float*)d_in[8];   // (224, 3072)
    const float* dt_proj_w  = (const float*)d_in[9];   // (3072, 96)
    const float* dt_proj_b  = (const float*)d_in[10];
    const float* Dp         = (const float*)d_in[12];
    const float* out_proj_w = (const float*)d_in[13];  // (1536, 3072)
    const float* norm2_g    = (const float*)d_in[14];
    const float* norm2_b    = (const float*)d_in[15];
    const float* cw1        = (const float*)d_in[16];
    const float* cw2        = (const float*)d_in[17];
    const float* cb1        = (const float*)d_in[18];
    const float* cb2        = (const float*)d_in[19];
    float* out = (float*)d_out;

    // workspace partition (floats)
    float* w = (float*)d_ws;
    float* ln1   = w; w += BATCH * DIMC;
    float* xz    = w; w += BATCH * 2 * DINNER;
    float* xc    = w; w += BATCH * DINNER;
    float* dbl   = w; w += BATCH * (DTRANK + 2 * DSTATE);
    float* dt    = w; w += BATCH * DINNER;
    float* dotBC = w; w += BATCH;
    float* yact  = w; w += BATCH * DINNER;
    float* cls1  = w; w += BATCH * DIMC;
    float* ln2   = w; w += BATCH * DIMC;
    float* Xr    = w; w += BATCH * DIMC;
    float* Xi    = w; w += BATCH * DIMC;
    float* r1    = w; w += BATCH * DIMC;
    float* i1    = w; w += BATCH * DIMC;
    float* r2    = w; w += BATCH * DIMC;
    float* i2    = w; w += BATCH * DIMC;

    const dim3 wave(32);
    const long xrow = (long)SEQ * DIMC;

    // 1) bandwidth-dominant pass-through copy of tokens 1..1024
    copy_tail_k<<<8192, 256, 0, stream>>>((const float4*)x, (float4*)out);

    // 2) Mamba branch on the CLS token
    layernorm_k<<<BATCH, 256, 0, stream>>>(x, xrow, norm1_g, norm1_b, ln1);
    gemm_wT_k<<<dim3(2 * DINNER / 16, BATCH / 16), wave, 0, stream>>>(
        ln1, DIMC, in_proj_w, xz, 2 * DINNER, DIMC, 0, nullptr, nullptr, 0);
    mid_k<<<(BATCH * DINNER + 255) / 256, 256, 0, stream>>>(xz, conv_w, conv_b, xc);
    gemm_wT_k<<<dim3((DTRANK + 2 * DSTATE) / 16, BATCH / 16), wave, 0, stream>>>(
        xc, DINNER, x_proj_w, dbl, DTRANK + 2 * DSTATE, DINNER, 0, nullptr, nullptr, 0);
    bc_k<<<BATCH, DSTATE, 0, stream>>>(dbl, dotBC);
    gemm_wT_k<<<dim3(DINNER / 16, BATCH / 16), wave, 0, stream>>>(
        dbl, DTRANK + 2 * DSTATE, dt_proj_w, dt, DINNER, DTRANK, 1, dt_proj_b, nullptr, 0);
    y_k<<<(BATCH * DINNER + 255) / 256, 256, 0, stream>>>(dt, dotBC, Dp, xc, xz, yact);
    gemm_wT_k<<<dim3(DIMC / 16, BATCH / 16), wave, 0, stream>>>(
        yact, DINNER, out_proj_w, cls1, DIMC, DINNER, 2, nullptr, x, xrow);

    // 3) EinFFT branch
    layernorm_k<<<BATCH, 256, 0, stream>>>(cls1, DIMC, norm2_g, norm2_b, ln2);
    fftpre_k<<<(BATCH * BLKC + 255) / 256, 256, 0, stream>>>(ln2, Xr, Xi);
    cgemm_k<<<dim3(BLKC / 16, BATCH / 16, NBLK), wave, 0, stream>>>(
        Xr, Xi, cw1, cb1, r1, i1, 0);
    cgemm_k<<<dim3(BLKC / 16, BATCH / 16, NBLK), wave, 0, stream>>>(
        r1, i1, cw2, cb2, r2, i2, 1);
    fftpost_k<<<(BATCH * BLKC + 255) / 256, 256, 0, stream>>>(r2, i2, cls1, out);
}